// OneToOne_1108101563130
// MI455X (gfx1250) — compile-verified
//
#include <hip/hip_runtime.h>
#include <hip/hip_bf16.h>

// out[t, f] = x[t, f] * w[f] + b[f]
// HBM-bandwidth-bound streaming kernel (~268 MB moved, ~11.5 us floor at
// 23.3 TB/s). Design goals this round:
//   - hot loop: 1x global_load_b128 (NT) + 2-4 VALU fma + 1x global_store_b128
//     (NT) per row, with the row stride folded into the 24-bit instruction
//     immediate offset (COLS4 is a template constant for the common shape)
//   - single uniform bounds branch per block (no per-row checks)
//   - pure 32-bit index math + scale_offset GVS addressing

typedef float __attribute__((ext_vector_type(4))) f32x4;

#define ROWS_PER_BLOCK 16
#define THREADS 256

template <int COLS4>
__global__ __launch_bounds__(THREADS) void onetoone_stream_fixed(
    const f32x4* __restrict__ x,
    const f32x4* __restrict__ w,
    const f32x4* __restrict__ b,
    f32x4* __restrict__ out,
    int rows)
{
    // grid.x * THREADS == COLS4 exactly (launcher guarantees it), no col guard.
    const unsigned col = blockIdx.x * THREADS + threadIdx.x;

    // weight/bias: 16 KiB each, reused by all rows -> regular-temporal,
    // stays resident in WGP$/L2 for the whole kernel.
    const f32x4 wv = w[col];
    const f32x4 bv = b[col];

    const int row0 = (int)blockIdx.y * ROWS_PER_BLOCK;
    const unsigned base = (unsigned)row0 * (unsigned)COLS4 + col;

    if (row0 + ROWS_PER_BLOCK <= rows) {
        // Fast path: branch-free, fully unrolled; all 16 loads/stores share
        // one base address, per-row stride is an immediate (r*COLS4*16 B).
#pragma unroll
        for (int r = 0; r < ROWS_PER_BLOCK; ++r) {
            // Speculative streaming prefetch one block-tile ahead
            // (global_prefetch_b8; OOB tail addresses silently dropped).
            if ((r & 3) == 0) {
                __builtin_prefetch(
                    (const void*)(x + base + (unsigned)(ROWS_PER_BLOCK + r) * COLS4),
                    0, 0);
            }
            f32x4 xv = __builtin_nontemporal_load(x + base + (unsigned)r * COLS4);
            f32x4 ov = xv * wv + bv;   // contracts to v_fma_f32 / v_pk_fma_f32
            __builtin_nontemporal_store(ov, out + base + (unsigned)r * COLS4);
        }
    } else {
        // Tail blocks (only when rows % ROWS_PER_BLOCK != 0).
        for (int r = 0; r < ROWS_PER_BLOCK && (row0 + r) < rows; ++r) {
            f32x4 xv = __builtin_nontemporal_load(x + base + (unsigned)r * COLS4);
            f32x4 ov = xv * wv + bv;
            __builtin_nontemporal_store(ov, out + base + (unsigned)r * COLS4);
        }
    }
}

// Generic fallback for shapes whose column count isn't a known specialization.
__global__ __launch_bounds__(THREADS) void onetoone_stream_generic(
    const f32x4* __restrict__ x,
    const f32x4* __restrict__ w,
    const f32x4* __restrict__ b,
    f32x4* __restrict__ out,
    int cols4, int rows)
{
    const int col = blockIdx.x * THREADS + threadIdx.x;
    if (col >= cols4) return;
    const f32x4 wv = w[col];
    const f32x4 bv = b[col];

    const int row0 = (int)blockIdx.y * ROWS_PER_BLOCK;
    unsigned idx = (unsigned)row0 * (unsigned)cols4 + (unsigned)col;

    if (row0 + ROWS_PER_BLOCK <= rows) {
#pragma unroll
        for (int r = 0; r < ROWS_PER_BLOCK; ++r) {
            f32x4 xv = __builtin_nontemporal_load(x + idx);
            f32x4 ov = xv * wv + bv;
            __builtin_nontemporal_store(ov, out + idx);
            idx += (unsigned)cols4;
        }
    } else {
        for (int r = 0; r < ROWS_PER_BLOCK && (row0 + r) < rows; ++r) {
            f32x4 xv = __builtin_nontemporal_load(x + idx);
            f32x4 ov = xv * wv + bv;
            __builtin_nontemporal_store(ov, out + idx);
            idx += (unsigned)cols4;
        }
    }
}

extern "C" void kernel_launch(void* const* d_in, const int* in_sizes, int n_in,
                              void* d_out, int out_size, void* d_ws, size_t ws_size,
                              hipStream_t stream) {
    (void)n_in; (void)out_size; (void)d_ws; (void)ws_size;

    const f32x4* x = (const f32x4*)d_in[0];   // (TOKENS, IN_FEATURES) f32
    const f32x4* w = (const f32x4*)d_in[1];   // (IN_FEATURES,) f32
    const f32x4* b = (const f32x4*)d_in[2];   // (IN_FEATURES,) f32
    f32x4* o = (f32x4*)d_out;

    const int in_features = in_sizes[1];                 // 4096
    const int tokens      = in_sizes[0] / in_features;   // 8192
    const int cols4       = in_features / 4;             // 1024

    const int grid_y = (tokens + ROWS_PER_BLOCK - 1) / ROWS_PER_BLOCK;

    if (cols4 == 1024) {
        dim3 grid(1024 / THREADS, grid_y, 1);            // (4, 512)
        onetoone_stream_fixed<1024><<<grid, dim3(THREADS), 0, stream>>>(
            x, w, b, o, tokens);
    } else {
        dim3 grid((cols4 + THREADS - 1) / THREADS, grid_y, 1);
        onetoone_stream_generic<<<grid, dim3(THREADS), 0, stream>>>(
            x, w, b, o, cols4, tokens);
    }
}